// Feature2Vertex_pytorch_25555055411674
// MI455X (gfx1250) — compile-verified
//
#include <hip/hip_runtime.h>
#include <math.h>

#define B_DIM  16
#define P_DIM  10000
#define NB_DIM 10
#define NCOL   48            // B_DIM * 3
#define KSPLIT 10            // waves per block in GEMM
#define KPW    (P_DIM / KSPLIT)   // 1000 K per wave
#define KSTEPS (KPW / 4)          // 250 wmma steps

typedef __attribute__((ext_vector_type(2))) float v2f;
typedef __attribute__((ext_vector_type(8))) float v8f;

// ---------------------------------------------------------------------------
// Kernel 1: T[b,p] = R(logr) @ S(sym)   (Rodrigues exactly as reference)
// ---------------------------------------------------------------------------
__global__ __launch_bounds__(256) void ftoT_kernel(const float* __restrict__ geo,
                                                   float* __restrict__ Tws) {
    int tid = blockIdx.x * blockDim.x + threadIdx.x;
    if (tid >= B_DIM * P_DIM) return;
    const float* f = geo + (size_t)tid * 9;
    float x = f[0], y = f[1], z = f[2];
    float theta = sqrtf(x * x + y * y + z * z);

    float R[9];
    if (theta == 0.0f) {
        R[0] = 1.f; R[1] = 0.f; R[2] = 0.f;
        R[3] = 0.f; R[4] = 1.f; R[5] = 0.f;
        R[6] = 0.f; R[7] = 0.f; R[8] = 1.f;
    } else {
        float inv = 1.0f / theta;
        float a = x * inv, b = y * inv, c = z * inv;
        float Kn[9] = { 0.f,  a,   b,
                       -a,   0.f,  c,
                       -b,  -c,   0.f };
        float K2[9];
#pragma unroll
        for (int i = 0; i < 3; ++i)
#pragma unroll
            for (int j = 0; j < 3; ++j)
                K2[i * 3 + j] = Kn[i * 3 + 0] * Kn[0 * 3 + j] +
                                Kn[i * 3 + 1] * Kn[1 * 3 + j] +
                                Kn[i * 3 + 2] * Kn[2 * 3 + j];
        float st = sinf(theta);
        float omc = 1.0f - cosf(theta);
#pragma unroll
        for (int i = 0; i < 9; ++i)
            R[i] = Kn[i] * st + K2[i] * omc;
        R[0] += 1.f; R[4] += 1.f; R[8] += 1.f;
    }

    float s3 = f[3], s4 = f[4], s5 = f[5], s6 = f[6], s7 = f[7], s8 = f[8];
    float S[9] = { s3, s4, s5,
                   s4, s6, s7,
                   s5, s7, s8 };
    float* T = Tws + (size_t)tid * 9;
#pragma unroll
    for (int i = 0; i < 3; ++i)
#pragma unroll
        for (int j = 0; j < 3; ++j)
            T[i * 3 + j] = R[i * 3 + 0] * S[0 * 3 + j] +
                           R[i * 3 + 1] * S[1 * 3 + j] +
                           R[i * 3 + 2] * S[2 * 3 + j];
}

// ---------------------------------------------------------------------------
// Kernel 2: bvecC[col][p] = sum_n (T[b, nb[p,n]-1 | 0] + T[b,p]) @ vdiff[p,n]
//           stored transposed (col = b*3+e) so the GEMM B-operand is
//           contiguous per lane (b64 loads).
// ---------------------------------------------------------------------------
__global__ __launch_bounds__(256) void bvec_kernel(const float* __restrict__ Tws,
                                                   const int*   __restrict__ nb,
                                                   const float* __restrict__ vdiff,
                                                   float* __restrict__ bvecC) {
    int tid = blockIdx.x * blockDim.x + threadIdx.x;
    if (tid >= B_DIM * P_DIM) return;
    int b = tid / P_DIM;
    int p = tid - b * P_DIM;

    const float* Tp = Tws + (size_t)tid * 9;
    float Tl[9];
#pragma unroll
    for (int i = 0; i < 9; ++i) Tl[i] = Tp[i];

    float acc0 = 0.f, acc1 = 0.f, acc2 = 0.f;
#pragma unroll
    for (int n = 0; n < NB_DIM; ++n) {
        int idx = nb[(size_t)p * NB_DIM + n];     // 0..P  (0 == zero-padded row)
        float Tn[9];
        if (idx == 0) {
#pragma unroll
            for (int i = 0; i < 9; ++i) Tn[i] = 0.f;
        } else {
            const float* Tq = Tws + ((size_t)b * P_DIM + (idx - 1)) * 9;
#pragma unroll
            for (int i = 0; i < 9; ++i) Tn[i] = Tq[i];
        }
        const float* vd = vdiff + ((size_t)p * NB_DIM + n) * 3;
        float v0 = vd[0], v1 = vd[1], v2 = vd[2];
        acc0 += (Tl[0] + Tn[0]) * v0 + (Tl[1] + Tn[1]) * v1 + (Tl[2] + Tn[2]) * v2;
        acc1 += (Tl[3] + Tn[3]) * v0 + (Tl[4] + Tn[4]) * v1 + (Tl[5] + Tn[5]) * v2;
        acc2 += (Tl[6] + Tn[6]) * v0 + (Tl[7] + Tn[7]) * v1 + (Tl[8] + Tn[8]) * v2;
    }
    bvecC[(size_t)(b * 3 + 0) * P_DIM + p] = acc0;
    bvecC[(size_t)(b * 3 + 1) * P_DIM + p] = acc1;
    bvecC[(size_t)(b * 3 + 2) * P_DIM + p] = acc2;
}

// ---------------------------------------------------------------------------
// Kernel 3: v[col][vtx] = sum_p recon[vtx][p] * bvecC[col][p]
//   D(16x16) = A(16x4) x B(4x16) via V_WMMA_F32_16X16X4_F32, split-K over
//   10 waves per block, deterministic LDS reduction.
//   A operand layout (32-bit 16x4): lane<16 holds row=lane K={k0,k0+1};
//   lane>=16 holds row=lane-16 K={k0+2,k0+3}  -> one b64 load per step.
//   B operand is the transpose of that layout; with bvecC stored [col][p]
//   each lane again loads a contiguous float2.
// ---------------------------------------------------------------------------
__global__ __launch_bounds__(KSPLIT * 32) void gemm_kernel(const float* __restrict__ A,
                                                           const float* __restrict__ Bm,
                                                           float* __restrict__ out) {
    const int mblk = blockIdx.x;              // 0..624  (16 vertex rows each)
    const int wave = threadIdx.x >> 5;        // 0..9
    const int lane = threadIdx.x & 31;
    const int l16  = lane & 15;
    const int hi   = lane >> 4;               // 0/1 -> K sub-offset 0/2
    const int k0   = wave * KPW;

    const float* ap  = A  + (size_t)(mblk * 16 + l16) * P_DIM + k0 + hi * 2;
    const float* bp0 = Bm + (size_t)(0  + l16) * P_DIM + k0 + hi * 2;
    const float* bp1 = Bm + (size_t)(16 + l16) * P_DIM + k0 + hi * 2;
    const float* bp2 = Bm + (size_t)(32 + l16) * P_DIM + k0 + hi * 2;

    v8f acc0 = {};
    v8f acc1 = {};
    v8f acc2 = {};

#pragma unroll 4
    for (int k = 0; k < KSTEPS; ++k) {
        v2f a  = *(const v2f*)ap;
        v2f b0 = *(const v2f*)bp0;
        v2f b1 = *(const v2f*)bp1;
        v2f b2 = *(const v2f*)bp2;
        ap += 4; bp0 += 4; bp1 += 4; bp2 += 4;
        acc0 = __builtin_amdgcn_wmma_f32_16x16x4_f32(false, a, false, b0,
                                                     (short)0, acc0, false, false);
        acc1 = __builtin_amdgcn_wmma_f32_16x16x4_f32(false, a, false, b1,
                                                     (short)0, acc1, false, false);
        acc2 = __builtin_amdgcn_wmma_f32_16x16x4_f32(false, a, false, b2,
                                                     (short)0, acc2, false, false);
    }

    // Deterministic cross-wave K reduction in LDS (fixed summation order).
    __shared__ float red[KSPLIT][32][24];
    float* slot = &red[wave][lane][0];
#pragma unroll
    for (int r = 0; r < 8; ++r) {
        slot[r]      = acc0[r];
        slot[8 + r]  = acc1[r];
        slot[16 + r] = acc2[r];
    }
    __syncthreads();

    if (wave == 0) {
        float s[24];
#pragma unroll
        for (int j = 0; j < 24; ++j) s[j] = red[0][lane][j];
        for (int w = 1; w < KSPLIT; ++w)
#pragma unroll
            for (int j = 0; j < 24; ++j) s[j] += red[w][lane][j];

        // C/D layout: VGPR r, lane<16 -> (M=r, N=lane); lane>=16 -> (M=r+8, N=lane-16)
#pragma unroll
        for (int t = 0; t < 3; ++t) {
            int col = t * 16 + l16;           // 0..47 = b*3 + e
            int bb  = col / 3;
            int e   = col - bb * 3;
#pragma unroll
            for (int r = 0; r < 8; ++r) {
                int vtx = mblk * 16 + r + hi * 8;
                out[((size_t)bb * P_DIM + vtx) * 3 + e] = s[t * 8 + r];
            }
        }
    }
}

// ---------------------------------------------------------------------------
extern "C" void kernel_launch(void* const* d_in, const int* in_sizes, int n_in,
                              void* d_out, int out_size, void* d_ws, size_t ws_size,
                              hipStream_t stream) {
    const float* geo   = (const float*)d_in[0];   // (B,P,9)  f32
    const int*   nb    = (const int*)  d_in[1];   // (P,NB)   i32
    const float* recon = (const float*)d_in[2];   // (P,P)    f32
    const float* vdiff = (const float*)d_in[3];   // (P,NB,3) f32
    float*       out   = (float*)d_out;           // (B,P,3)  f32

    float* Tws   = (float*)d_ws;                              // B*P*9   floats
    float* bvecC = Tws + (size_t)B_DIM * P_DIM * 9;           // NCOL*P  floats

    const int nbp    = B_DIM * P_DIM;                          // 160000
    const int blocks = (nbp + 255) / 256;                      // 625

    ftoT_kernel<<<blocks, 256, 0, stream>>>(geo, Tws);
    bvec_kernel<<<blocks, 256, 0, stream>>>(Tws, nb, vdiff, bvecC);
    gemm_kernel<<<P_DIM / 16, KSPLIT * 32, 0, stream>>>(recon, bvecC, out);
}